// SMIL_41970420417206
// MI455X (gfx1250) — compile-verified
//
#include <hip/hip_runtime.h>
#include <cstdint>

#define Bn   1024
#define Kj   24
#define Vn   6890
#define V3   (Vn*3)      // 20670 valid rows
#define V3S  20672       // padded per-batch stride for v_shaped (mult of 4 floats)
#define V3A  20736       // padded A rows = 81*16 tiles * 16
#define Lk   207
#define LP   208         // K padded to multiple of 4

typedef __attribute__((ext_vector_type(2))) float v2f;
typedef __attribute__((ext_vector_type(8))) float v8f;

__constant__ int c_parents[Kj] = {-1,0,0,0,1,2,3,4,5,6,7,8,9,9,9,12,13,14,16,17,18,19,20,21};

// ---------------- 1) pad posedirs [V,3,207] -> [V3A, 208] (zero padded) ----
__global__ void k_pad_posedirs(const float* __restrict__ pd, float* __restrict__ out) {
    size_t i = (size_t)blockIdx.x * 256 + threadIdx.x;   // over V3A*LP
    int row = (int)(i / LP);
    int col = (int)(i % LP);
    float v = 0.0f;
    if (row < V3 && col < Lk) v = pd[(size_t)row * Lk + col];
    out[i] = v;
}

// ---------------- 2) rodrigues -> Rcube[B,24,9], lrotT[208,B] --------------
__global__ void k_pose(const float* __restrict__ pose,
                       float* __restrict__ Rcube, float* __restrict__ lrotT) {
    int t = blockIdx.x * blockDim.x + threadIdx.x;       // B*K threads
    if (t >= Bn * Kj) return;
    int b = t / Kj, k = t % Kj;
    float rx = pose[b*72 + k*3 + 0];
    float ry = pose[b*72 + k*3 + 1];
    float rz = pose[b*72 + k*3 + 2];
    float th = sqrtf(rx*rx + ry*ry + rz*rz);
    th = fmaxf(th, 1.1754944e-38f);
    float inv = 1.0f / th;
    float x = rx*inv, y = ry*inv, z = rz*inv;
    float s = sinf(th), c = cosf(th);
    float omc = 1.0f - c;
    float R[9];
    R[0] = c + omc*x*x;   R[1] = omc*x*y - s*z; R[2] = omc*x*z + s*y;
    R[3] = omc*y*x + s*z; R[4] = c + omc*y*y;   R[5] = omc*y*z - s*x;
    R[6] = omc*z*x - s*y; R[7] = omc*z*y + s*x; R[8] = c + omc*z*z;
#pragma unroll
    for (int e = 0; e < 9; e++) Rcube[((size_t)b*Kj + k)*9 + e] = R[e];
    if (k >= 1) {
        int base = (k - 1) * 9;
#pragma unroll
        for (int e = 0; e < 9; e++)
            lrotT[(size_t)(base + e)*Bn + b] = R[e] - ((e % 4 == 0) ? 1.0f : 0.0f);
    } else {
        lrotT[(size_t)207*Bn + b] = 0.0f;  // zero pad row
    }
}

// ---------------- 3) v_shaped = shapedirs.beta + v_template ----------------
__global__ void k_vshaped(const float* __restrict__ beta,
                          const float* __restrict__ shapedirs,
                          const float* __restrict__ v_template,
                          float* __restrict__ vsh) {
    int m = blockIdx.x * 256 + threadIdx.x;
    int b = blockIdx.y;
    if (m >= V3S) return;
    float acc = 0.0f;
    if (m < V3) {
        const float* bet = beta + b*10;
        const float* sd  = shapedirs + (size_t)m*10;
        acc = v_template[m];
#pragma unroll
        for (int l = 0; l < 10; l++) acc = fmaf(bet[l], sd[l], acc);
    }
    vsh[(size_t)b*V3S + m] = acc;
}

// ---------------- 4) J = J_regressor . v_shaped ----------------------------
__global__ void k_J(const float* __restrict__ Jreg, const float* __restrict__ vsh,
                    float* __restrict__ Jout) {
    int blk = blockIdx.x;
    int b = blk / Kj, k = blk % Kj;
    const float* jr = Jreg + (size_t)k*Vn;
    const float* vs = vsh + (size_t)b*V3S;
    float a0 = 0.f, a1 = 0.f, a2 = 0.f;
    for (int v = threadIdx.x; v < Vn; v += 64) {
        float w = jr[v];
        a0 = fmaf(w, vs[v*3+0], a0);
        a1 = fmaf(w, vs[v*3+1], a1);
        a2 = fmaf(w, vs[v*3+2], a2);
    }
    __shared__ float red[3][64];
    red[0][threadIdx.x] = a0; red[1][threadIdx.x] = a1; red[2][threadIdx.x] = a2;
    __syncthreads();
    for (int s = 32; s > 0; s >>= 1) {
        if ((int)threadIdx.x < s) {
            red[0][threadIdx.x] += red[0][threadIdx.x + s];
            red[1][threadIdx.x] += red[1][threadIdx.x + s];
            red[2][threadIdx.x] += red[2][threadIdx.x + s];
        }
        __syncthreads();
    }
    if (threadIdx.x == 0) {
        Jout[((size_t)b*Kj + k)*3 + 0] = red[0][0];
        Jout[((size_t)b*Kj + k)*3 + 1] = red[1][0];
        Jout[((size_t)b*Kj + k)*3 + 2] = red[2][0];
    }
}

// ---------------- 5) kinematic chain -> Grel[B,24,12], Jtr -----------------
__global__ void __launch_bounds__(32) k_chain(const float* __restrict__ Rcube,
                                              const float* __restrict__ Jbuf,
                                              const float* __restrict__ trans,
                                              float* __restrict__ Grel,
                                              float* __restrict__ JtrOut) {
    __shared__ float Gs[32][Kj][12];
    int b = blockIdx.x * 32 + threadIdx.x;   // grid 32 x 32 = 1024
    float (*G)[12] = Gs[threadIdx.x];
    const float* R0 = Rcube + (size_t)b*Kj*9;
    const float* J0 = Jbuf  + (size_t)b*Kj*3;
#pragma unroll
    for (int i = 0; i < 3; i++) {
        G[0][i*4+0] = R0[i*3+0]; G[0][i*4+1] = R0[i*3+1];
        G[0][i*4+2] = R0[i*3+2]; G[0][i*4+3] = J0[i];
    }
    for (int k = 1; k < Kj; k++) {
        int p = c_parents[k];
        const float* Rk = R0 + k*9;
        float A[12];
#pragma unroll
        for (int i = 0; i < 3; i++) {
            A[i*4+0] = Rk[i*3+0]; A[i*4+1] = Rk[i*3+1]; A[i*4+2] = Rk[i*3+2];
            A[i*4+3] = J0[k*3+i] - J0[p*3+i];
        }
        float* Gp = G[p];
        float* Gk = G[k];
#pragma unroll
        for (int i = 0; i < 3; i++) {
#pragma unroll
            for (int j = 0; j < 4; j++) {
                float v = Gp[i*4+0]*A[0*4+j] + Gp[i*4+1]*A[1*4+j] + Gp[i*4+2]*A[2*4+j];
                if (j == 3) v += Gp[i*4+3];
                Gk[i*4+j] = v;
            }
        }
    }
    float tr0 = trans[b*3+0], tr1 = trans[b*3+1], tr2 = trans[b*3+2];
    for (int k = 0; k < Kj; k++) {
        float* Gk = G[k];
        size_t jo = ((size_t)b*Kj + k)*4;
        JtrOut[jo+0] = Gk[3]  + tr0;
        JtrOut[jo+1] = Gk[7]  + tr1;
        JtrOut[jo+2] = Gk[11] + tr2;
        JtrOut[jo+3] = 1.0f;
        float Jx = J0[k*3+0], Jy = J0[k*3+1], Jz = J0[k*3+2];
        size_t go = ((size_t)b*Kj + k)*12;
#pragma unroll
        for (int i = 0; i < 3; i++) {
            float corr = Gk[i*4+0]*Jx + Gk[i*4+1]*Jy + Gk[i*4+2]*Jz;
            Grel[go + i*4+0] = Gk[i*4+0];
            Grel[go + i*4+1] = Gk[i*4+1];
            Grel[go + i*4+2] = Gk[i*4+2];
            Grel[go + i*4+3] = Gk[i*4+3] - corr;
        }
    }
}

// ---------------- 6) WMMA pose-blend GEMM: vsh += posedP x lrotT -----------
// 4 m-tiles (16x16 each) per wave, shared B-operand from LDS.
__global__ void __launch_bounds__(128) k_blend(const float* __restrict__ posedP,
                                               const float* __restrict__ lrotT,
                                               float* __restrict__ vsh) {
    __shared__ float Bs[LP * 16];            // 208 x 16 batch panel, 13.3 KB
    int b0 = blockIdx.y * 16;
    for (int i = threadIdx.x; i < LP * 16; i += 128) {
        int l = i >> 4, n = i & 15;
        Bs[i] = lrotT[(size_t)l * Bn + b0 + n];
    }
    __syncthreads();

    int wave  = threadIdx.x >> 5;
    int lane  = threadIdx.x & 31;
    int tile0 = (blockIdx.x * 4 + wave) * 4; // 4 consecutive m-tiles per wave
    int half  = lane >> 4;                   // 0: K0/K1 ; 1: K2/K3
    int mn    = lane & 15;                   // A row / B,D column index
    int c0    = half * 2;
    const float* ar = posedP + (size_t)(tile0 * 16 + mn) * LP;

    v8f acc0 = {}, acc1 = {}, acc2 = {}, acc3 = {};
    for (int kk = 0; kk < LP / 4; kk++) {    // 52 chunks of K=4
        int kb = kk * 4 + c0;
        v2f bb;
        bb.x = Bs[kb * 16 + mn];             // B[kb,   b0+mn]
        bb.y = Bs[(kb + 1) * 16 + mn];       // B[kb+1, b0+mn]
        v2f a0 = *(const v2f*)(ar + kb);
        v2f a1 = *(const v2f*)(ar + (size_t)16 * LP + kb);
        v2f a2 = *(const v2f*)(ar + (size_t)32 * LP + kb);
        v2f a3 = *(const v2f*)(ar + (size_t)48 * LP + kb);
        acc0 = __builtin_amdgcn_wmma_f32_16x16x4_f32(false, a0, false, bb, (short)0, acc0, false, false);
        acc1 = __builtin_amdgcn_wmma_f32_16x16x4_f32(false, a1, false, bb, (short)0, acc1, false, false);
        acc2 = __builtin_amdgcn_wmma_f32_16x16x4_f32(false, a2, false, bb, (short)0, acc2, false, false);
        acc3 = __builtin_amdgcn_wmma_f32_16x16x4_f32(false, a3, false, bb, (short)0, acc3, false, false);
    }

    // D layout: VGPR r -> M = r + 8*half, N = mn. Each lane's 8 elems are
    // contiguous in m; column base is 16B aligned (V3S % 4 == 0).
    size_t colBase = (size_t)(b0 + mn) * V3S;
#pragma unroll
    for (int t = 0; t < 4; t++) {
        int m0 = (tile0 + t) * 16;
        if (m0 >= V3S) break;                // fully padded tiles at the end
        v8f a = (t == 0) ? acc0 : (t == 1) ? acc1 : (t == 2) ? acc2 : acc3;
        float4* p = (float4*)(vsh + colBase + m0 + half * 8);
        float4 lo = p[0], hi = p[1];
        lo.x += a[0]; lo.y += a[1]; lo.z += a[2]; lo.w += a[3];
        hi.x += a[4]; hi.y += a[5]; hi.z += a[6]; hi.w += a[7];
        p[0] = lo; p[1] = hi;
    }
}

// ---------------- 7) skinning ----------------------------------------------
__global__ void __launch_bounds__(256) k_skin(const float* __restrict__ vsh,
                                              const float* __restrict__ weights,
                                              const float* __restrict__ Grel,
                                              const float* __restrict__ trans,
                                              float* __restrict__ outV) {
    __shared__ float Gls[Kj * 12];
    int b = blockIdx.y;
    for (int i = threadIdx.x; i < Kj * 12; i += 256)
        Gls[i] = Grel[(size_t)b * Kj * 12 + i];
    __syncthreads();
    int v = blockIdx.x * 256 + threadIdx.x;
    if (v >= Vn) return;
    const float* p = vsh + (size_t)b * V3S + (size_t)v * 3;
    float p0 = p[0], p1 = p[1], p2 = p[2];
    const float* wv = weights + (size_t)v * Kj;
    float o0 = 0.f, o1 = 0.f, o2 = 0.f, o3 = 0.f;
#pragma unroll
    for (int k = 0; k < Kj; k++) {
        float w = wv[k];
        const float* g = Gls + k * 12;
        o0 = fmaf(w, fmaf(g[0], p0, fmaf(g[1], p1, fmaf(g[2],  p2, g[3]))),  o0);
        o1 = fmaf(w, fmaf(g[4], p0, fmaf(g[5], p1, fmaf(g[6],  p2, g[7]))),  o1);
        o2 = fmaf(w, fmaf(g[8], p0, fmaf(g[9], p1, fmaf(g[10], p2, g[11]))), o2);
        o3 += w;
    }
    float4 out;
    out.x = o0 + trans[b*3+0];
    out.y = o1 + trans[b*3+1];
    out.z = o2 + trans[b*3+2];
    out.w = o3;
    *(float4*)(outV + ((size_t)b * Vn + v) * 4) = out;
}

// ---------------- launcher -------------------------------------------------
extern "C" void kernel_launch(void* const* d_in, const int* in_sizes, int n_in,
                              void* d_out, int out_size, void* d_ws, size_t ws_size,
                              hipStream_t stream) {
    const float* beta       = (const float*)d_in[0];  // [B,10]
    const float* pose       = (const float*)d_in[1];  // [B,72]
    const float* trans      = (const float*)d_in[2];  // [B,3]
    const float* weights    = (const float*)d_in[3];  // [V,24]
    const float* posedirs   = (const float*)d_in[4];  // [V,3,207]
    const float* v_template = (const float*)d_in[5];  // [V,3]
    const float* shapedirs  = (const float*)d_in[6];  // [V,3,10]
    const float* Jreg       = (const float*)d_in[7];  // [24,V]

    float* outV = (float*)d_out;                      // [B,V,4]
    float* outJ = outV + (size_t)Bn * Vn * 4;         // [B,24,4]

    float* w      = (float*)d_ws;
    float* Rcube  = w;                                // B*24*9   = 221184
    float* lrotT  = Rcube + (size_t)Bn*Kj*9;          // 208*B    = 212992
    float* vsh    = lrotT + (size_t)LP*Bn;            // B*V3S    (16B-aligned base)
    float* posedP = vsh + (size_t)Bn*V3S;             // V3A*208
    float* Jbuf   = posedP + (size_t)V3A*LP;          // B*24*3
    float* Grel   = Jbuf + (size_t)Bn*Kj*3;           // B*24*12

    // 1) pad posedirs (V3A*LP = 4,313,088 elems -> 16848 blocks)
    k_pad_posedirs<<<dim3((V3A*LP)/256), 256, 0, stream>>>(posedirs, posedP);
    // 2) rodrigues + lrotmin^T
    k_pose<<<dim3((Bn*Kj)/128), 128, 0, stream>>>(pose, Rcube, lrotT);
    // 3) v_shaped (writes padded stride, zeros in pad)
    k_vshaped<<<dim3(V3S/256 + 1, Bn), 256, 0, stream>>>(beta, shapedirs, v_template, vsh);
    // 4) J regression (must precede in-place blend)
    k_J<<<dim3(Bn*Kj), 64, 0, stream>>>(Jreg, vsh, Jbuf);
    // 5) kinematic chain -> Grel + Jtr
    k_chain<<<dim3(Bn/32), 32, 0, stream>>>(Rcube, Jbuf, trans, Grel, outJ);
    // 6) WMMA pose-blend GEMM, in-place into vsh (81*16 tiles x 64 b-tiles)
    k_blend<<<dim3(81, Bn/16), 128, 0, stream>>>(posedP, lrotT, vsh);
    // 7) skinning
    k_skin<<<dim3((Vn + 255)/256, Bn), 256, 0, stream>>>(vsh, weights, Grel, trans, outV);
}